// padding_reshape_layer_62998580298150
// MI455X (gfx1250) — compile-verified
//
#include <hip/hip_runtime.h>
#include <hip/hip_bf16.h>
#include <stdint.h>

#define GRID14   14
#define MAX_SP   196          // 14*14
#define DIM      192
#define LDS_STRIDE 198        // 192 data + 6 pad dwords; 198%64=6 -> conflict-free q-strided reads
#define LDS_FLOATS (MAX_SP * LDS_STRIDE)   // 38808 floats = 155232 bytes (2 blocks / 320KB WGP)

// ---------------------------------------------------------------------------
// Kernel 1: exclusive prefix sum of segment counts -> offsets (B <= 1024)
// ---------------------------------------------------------------------------
__global__ __launch_bounds__(1024) void seg_offsets_kernel(
    const int* __restrict__ counts, int* __restrict__ offsets, int n) {
  __shared__ int tmp[1024];
  const int t = threadIdx.x;
  const int myc = (t < n) ? counts[t] : 0;
  tmp[t] = myc;
  __syncthreads();
  // Hillis-Steele inclusive scan over 1024 slots
  #pragma unroll
  for (int d = 1; d < 1024; d <<= 1) {
    const int v = (t >= d) ? tmp[t - d] : 0;
    __syncthreads();
    tmp[t] += v;
    __syncthreads();
  }
  if (t < n) offsets[t] = tmp[t] - myc;  // exclusive
}

// ---------------------------------------------------------------------------
// Kernel 2: per-image gather + pad + (c,hw) transpose via async-DMA LDS stage
//   out[b][c][j*14+i] = NF[off_b + i*14+j][c]  (0 if i*14+j >= count_b)
// ---------------------------------------------------------------------------
__global__ __launch_bounds__(256) void pad_permute_kernel(
    const float* __restrict__ nf,
    const int*   __restrict__ counts,
    const int*   __restrict__ offsets,
    float*       __restrict__ out) {
  extern __shared__ float smem[];   // [196][LDS_STRIDE], row q holds channels of seg row perm(q)

  const int b     = blockIdx.x;
  const int tid   = threadIdx.x;            // 256 threads = 8 waves
  const int count = counts[b];
  const int off   = offsets[b];

  const char* gbase = (const char*)(nf + (size_t)off * DIM);

  // ---- Stage: async DMA global -> LDS, 8B chunks, permutation folded into
  //      the per-lane LDS destination address. Row = 192 floats = 96 b64 chunks.
  const int nchunks = count * (DIM / 2 / 1);   // count * 96 b64 chunks
  for (int g = tid; g < nchunks; g += 256) {
    const int s  = g / 96;                 // segment row (w-major spatial index)
    const int c8 = g - s * 96;             // b64 chunk within row
    const int q  = (s % GRID14) * GRID14 + (s / GRID14);   // h/w-swapped row
    const unsigned ldsoff =
        (unsigned)(uintptr_t)(const void*)&smem[q * LDS_STRIDE + c8 * 2];
    const void* gp = (const void*)(gbase + (size_t)g * 8);
    asm volatile("global_load_async_to_lds_b64 %0, %1, off"
                 :
                 : "v"(ldsoff), "v"(gp)
                 : "memory");
  }
  asm volatile("s_wait_asynccnt 0x0" ::: "memory");
  __syncthreads();

  // ---- Drain: out[b] is c-major rows of 196; flat index g = c*196 + q.
  //      Lanes sweep q -> coalesced stores; LDS read stride 198 dwords -> conflict-free.
  float* obase = out + (size_t)b * (DIM * MAX_SP);
  for (int g = tid; g < DIM * MAX_SP; g += 256) {
    const int c = g / MAX_SP;
    const int q = g - c * MAX_SP;
    const int s = (q % GRID14) * GRID14 + (q / GRID14);    // inverse (involution)
    const float v = (s < count) ? smem[q * LDS_STRIDE + c] : 0.0f;
    __builtin_nontemporal_store(v, &obase[g]);
  }
}

// ---------------------------------------------------------------------------
extern "C" void kernel_launch(void* const* d_in, const int* in_sizes, int n_in,
                              void* d_out, int out_size, void* d_ws, size_t ws_size,
                              hipStream_t stream) {
  const float* nf     = (const float*)d_in[0];   // [total, 192] f32
  const int*   counts = (const int*)  d_in[1];   // [B] i32
  const int    B      = in_sizes[1];             // 1024

  int* offsets = (int*)d_ws;                     // B ints of scratch

  seg_offsets_kernel<<<1, 1024, 0, stream>>>(counts, offsets, B);

  const size_t shmem = (size_t)LDS_FLOATS * sizeof(float);   // 155232 B
  pad_permute_kernel<<<B, 256, shmem, stream>>>(nf, counts, offsets, (float*)d_out);
}